// GATLayer_75788992905478
// MI455X (gfx1250) — compile-verified
//
#include <hip/hip_runtime.h>
#include <hip/hip_bf16.h>

#define H 2
#define C 128
#define ED 16
#define INF 256          // input feature dim
#define HC 256           // H*C
#define SLOPE 0.2f

typedef __attribute__((ext_vector_type(2))) float v2f;
typedef __attribute__((ext_vector_type(8))) float v8f;

// ---- monotone float<->int key for atomic segment-max -----------------------
__device__ __forceinline__ int ordkey(float f) {
    int i = __float_as_int(f);
    return (i >= 0) ? i : (i ^ 0x7fffffff);
}
__device__ __forceinline__ float unord(int k) {
    return __int_as_float((k >= 0) ? k : (k ^ 0x7fffffff));
}

// ---- K0a: zero segment stats, init amax keys -------------------------------
__global__ void init_stats_kernel(float* sum_attr, float* cnt, int* amax,
                                  float* asum, int N) {
    int idx = blockIdx.x * blockDim.x + threadIdx.x;
    if (idx >= N * ED) return;
    sum_attr[idx] = 0.0f;
    if (idx < N) cnt[idx] = 0.0f;
    if (idx < N * H) { amax[idx] = (int)0x80000000; asum[idx] = 0.0f; }
}

// ---- K0b: out initialized to bias (aggregation accumulates on top) ---------
__global__ void init_out_kernel(float* out, const float* __restrict__ bias, int N) {
    int idx = blockIdx.x * blockDim.x + threadIdx.x;
    if (idx >= N * C) return;
    out[idx] = bias[idx & (C - 1)];
}

// ---- K1: segment-sum edge_attr over dst (for self-loop fill='mean') --------
__global__ void edge_attr_accum_kernel(const int* __restrict__ dstIdx,
                                       const float* __restrict__ edge_attr,
                                       float* sum_attr, float* cnt, int E) {
    int idx = blockIdx.x * blockDim.x + threadIdx.x;
    if (idx >= E * ED) return;
    int e = idx >> 4, d16 = idx & 15;
    int c = dstIdx[e];
    atomicAdd(&sum_attr[(size_t)c * ED + d16], edge_attr[idx]);
    if (d16 == 0) atomicAdd(&cnt[c], 1.0f);
}

// ---- K2: loop_attr = sum / max(cnt,1) --------------------------------------
__global__ void loop_attr_kernel(const float* __restrict__ sum_attr,
                                 const float* __restrict__ cnt,
                                 float* loop_attr, int N) {
    int idx = blockIdx.x * blockDim.x + threadIdx.x;
    if (idx >= N * ED) return;
    loop_attr[idx] = sum_attr[idx] / fmaxf(cnt[idx >> 4], 1.0f);
}

// ---- K3: vedge[d][h] = sum_c W_edge[d, h*C+c] * att_edge[h,c]  (16x2) ------
__global__ void vedge_kernel(const float* __restrict__ W_edge,
                             const float* __restrict__ att_edge, float* vedge) {
    int t = threadIdx.x;
    if (t >= ED * H) return;
    int d = t >> 1, h = t & 1;
    float s = 0.0f;
    for (int c = 0; c < C; ++c)
        s += W_edge[d * HC + h * C + c] * att_edge[h * C + c];
    vedge[d * H + h] = s;
}

// ---- K4: xh = x @ W via V_WMMA_F32_16X16X4_F32 (full fp32 matrix pipe) -----
#define LDSS 264   // padded row stride (floats) to spread LDS banks
__global__ __launch_bounds__(512) void gemm_xh_kernel(const float* __restrict__ x,
                                                      const float* __restrict__ W,
                                                      float* __restrict__ xh) {
    __shared__ float lds_x[16 * LDSS];
    const int row0 = blockIdx.x * 16;
    const int tid = threadIdx.x;

    // cooperative load of 16x256 fp32 x-tile (float4, coalesced)
    for (int i = tid; i < 16 * 64; i += 512) {
        int r = i >> 6;          // row 0..15
        int c4 = i & 63;         // float4 index 0..63
        float4 v = ((const float4*)(x + (size_t)(row0 + r) * INF))[c4];
        *(float4*)&lds_x[r * LDSS + c4 * 4] = v;
    }
    __syncthreads();

    const int wave = tid >> 5;       // 16 waves -> 16-col strips
    const int lane = tid & 31;
    const int lg = lane >> 4;        // lane group: K pair select
    const int lm = lane & 15;        // M (for A) / N (for B,D)
    const int col0 = wave * 16;

    v8f acc = {};
    for (int k = 0; k < INF; k += 4) {
        const int ka = k + 2 * lg;
        v2f a, b;
        a.x = lds_x[lm * LDSS + ka];
        a.y = lds_x[lm * LDSS + ka + 1];
        b.x = W[(size_t)ka * HC + col0 + lm];
        b.y = W[(size_t)(ka + 1) * HC + col0 + lm];
        acc = __builtin_amdgcn_wmma_f32_16x16x4_f32(
            /*neg_a=*/false, a, /*neg_b=*/false, b,
            /*c_mod=*/(short)0, acc, /*reuse_a=*/false, /*reuse_b=*/false);
    }
    // D layout: VGPR i holds row i (lanes 0-15) / row 8+i (lanes 16-31), col=lm
#pragma unroll
    for (int i = 0; i < 8; ++i) {
        int m = i + 8 * lg;
        xh[(size_t)(row0 + m) * HC + col0 + lm] = acc[i];
    }
}

// ---- K5: a_src/a_dst = per-(n,h) projections of xh; one wave per pair ------
__global__ __launch_bounds__(256) void node_att_kernel(const float* __restrict__ xh,
                                                       const float* __restrict__ att_src,
                                                       const float* __restrict__ att_dst,
                                                       float* a_src, float* a_dst, int N) {
    int pair = (blockIdx.x * blockDim.x + threadIdx.x) >> 5;
    int lane = threadIdx.x & 31;
    if (pair >= N * H) return;
    int n = pair >> 1, h = pair & 1;
    const float* row = xh + (size_t)n * HC + h * C;
    const float* as = att_src + h * C;
    const float* ad = att_dst + h * C;
    float ss = 0.0f, sd = 0.0f;
#pragma unroll
    for (int j = 0; j < 4; ++j) {
        float v = row[lane + 32 * j];
        ss += v * as[lane + 32 * j];
        sd += v * ad[lane + 32 * j];
    }
#pragma unroll
    for (int off = 16; off; off >>= 1) {
        ss += __shfl_down(ss, off, 32);
        sd += __shfl_down(sd, off, 32);
    }
    if (lane == 0) { a_src[pair] = ss; a_dst[pair] = sd; }
}

// ---- K6: raw logits + leaky_relu + atomic segment-max; also emit ei --------
__global__ void edge_logits_kernel(const int* __restrict__ srcIdx,
                                   const int* __restrict__ dstIdx,
                                   const float* __restrict__ edge_attr,
                                   const float* __restrict__ loop_attr,
                                   const float* __restrict__ vedge,
                                   const float* __restrict__ a_src,
                                   const float* __restrict__ a_dst,
                                   float* raw, int* amax, float* ei, int E, int N) {
    int e = blockIdx.x * blockDim.x + threadIdx.x;
    if (e >= E + N) return;
    int s, d;
    const float* ea;
    if (e < E) { s = srcIdx[e]; d = dstIdx[e]; ea = edge_attr + (size_t)e * ED; }
    else       { s = e - E;     d = s;         ea = loop_attr + (size_t)s * ED; }

    float ae0 = 0.0f, ae1 = 0.0f;
#pragma unroll
    for (int i = 0; i < ED; ++i) {
        float v = ea[i];
        ae0 += v * vedge[i * H + 0];
        ae1 += v * vedge[i * H + 1];
    }
#pragma unroll
    for (int h = 0; h < H; ++h) {
        float al = a_src[s * H + h] + a_dst[d * H + h] + (h ? ae1 : ae0);
        al = (al > 0.0f) ? al : SLOPE * al;            // leaky_relu
        raw[(size_t)e * H + h] = al;
        atomicMax(&amax[d * H + h], ordkey(al));       // segment max
    }
    ei[e] = (float)s;
    ei[(size_t)(E + N) + e] = (float)d;
}

// ---- K7: exp(raw - amax[dst]) -> alpha(out buf); atomic segment-sum --------
__global__ void edge_exp_kernel(const int* __restrict__ dstIdx,
                                const float* __restrict__ raw,
                                const int* __restrict__ amax,
                                float* asum, float* alpha, int E, int N) {
    int idx = blockIdx.x * blockDim.x + threadIdx.x;
    if (idx >= (E + N) * H) return;
    int e = idx >> 1, h = idx & 1;
    int d = (e < E) ? dstIdx[e] : (e - E);
    float m = unord(amax[d * H + h]);
    float v = __expf(raw[idx] - m);
    alpha[idx] = v;
    atomicAdd(&asum[d * H + h], v);
}

// ---- K8: normalize alpha in place; wave/edge gather+scatter aggregation ----
__global__ __launch_bounds__(256) void aggregate_kernel(const int* __restrict__ srcIdx,
                                                        const int* __restrict__ dstIdx,
                                                        const float* __restrict__ xh,
                                                        const float* __restrict__ asum,
                                                        float* alpha, float* out,
                                                        int E, int N) {
    int gw = (blockIdx.x * blockDim.x + threadIdx.x) >> 5;
    int lane = threadIdx.x & 31;
    if (gw >= E + N) return;
    int s, d;
    if (gw < E) { s = srcIdx[gw]; d = dstIdx[gw]; } else { s = gw - E; d = s; }

    float e0 = alpha[(size_t)gw * H + 0];
    float e1 = alpha[(size_t)gw * H + 1];
    float a0 = e0 / (asum[d * H + 0] + 1e-16f);
    float a1 = e1 / (asum[d * H + 1] + 1e-16f);
    if (lane == 0) {                      // write normalized alpha (output)
        alpha[(size_t)gw * H + 0] = a0;
        alpha[(size_t)gw * H + 1] = a1;
    }
    // head-mean folded in: out[d,c] += 0.5*(a0*xh[s,0,c] + a1*xh[s,1,c])
    const float4 x0 = *(const float4*)(xh + (size_t)s * HC + lane * 4);
    const float4 x1 = *(const float4*)(xh + (size_t)s * HC + C + lane * 4);
    float4 cb;
    cb.x = 0.5f * (a0 * x0.x + a1 * x1.x);
    cb.y = 0.5f * (a0 * x0.y + a1 * x1.y);
    cb.z = 0.5f * (a0 * x0.z + a1 * x1.z);
    cb.w = 0.5f * (a0 * x0.w + a1 * x1.w);
    float* dp = out + (size_t)d * C + lane * 4;
    atomicAdd(dp + 0, cb.x);
    atomicAdd(dp + 1, cb.y);
    atomicAdd(dp + 2, cb.z);
    atomicAdd(dp + 3, cb.w);
}

extern "C" void kernel_launch(void* const* d_in, const int* in_sizes, int n_in,
                              void* d_out, int out_size, void* d_ws, size_t ws_size,
                              hipStream_t stream) {
    const float* x        = (const float*)d_in[0];
    const int*   eidx     = (const int*)d_in[1];
    const float* edge_attr= (const float*)d_in[2];
    const float* W        = (const float*)d_in[3];
    const float* W_edge   = (const float*)d_in[4];
    const float* att_src  = (const float*)d_in[5];
    const float* att_dst  = (const float*)d_in[6];
    const float* att_edge = (const float*)d_in[7];
    const float* bias     = (const float*)d_in[8];

    const int N = in_sizes[0] / INF;
    const int E = in_sizes[1] / 2;
    const int* srcIdx = eidx;
    const int* dstIdx = eidx + E;

    // workspace layout (floats)
    float* ws = (float*)d_ws;
    size_t o = 0;
    float* xh       = ws + o; o += (size_t)N * HC;
    float* sum_attr = ws + o; o += (size_t)N * ED;
    float* cnt      = ws + o; o += (size_t)N;
    float* loop_at  = ws + o; o += (size_t)N * ED;
    float* a_src    = ws + o; o += (size_t)N * H;
    float* a_dst    = ws + o; o += (size_t)N * H;
    int*   amax     = (int*)(ws + o); o += (size_t)N * H;
    float* asum     = ws + o; o += (size_t)N * H;
    float* raw      = ws + o; o += (size_t)(E + N) * H;
    float* vedge    = ws + o; o += ED * H;

    // output layout: out [N*C] | ei [2*(E+N)] | alpha [(E+N)*H]
    float* out   = (float*)d_out;
    float* ei    = out + (size_t)N * C;
    float* alpha = ei + 2 * (size_t)(E + N);

    const int BT = 256;
    init_stats_kernel<<<(N * ED + BT - 1) / BT, BT, 0, stream>>>(sum_attr, cnt, amax, asum, N);
    init_out_kernel<<<(N * C + BT - 1) / BT, BT, 0, stream>>>(out, bias, N);
    edge_attr_accum_kernel<<<((size_t)E * ED + BT - 1) / BT, BT, 0, stream>>>(
        dstIdx, edge_attr, sum_attr, cnt, E);
    loop_attr_kernel<<<(N * ED + BT - 1) / BT, BT, 0, stream>>>(sum_attr, cnt, loop_at, N);
    vedge_kernel<<<1, 32, 0, stream>>>(W_edge, att_edge, vedge);

    gemm_xh_kernel<<<N / 16, 512, 0, stream>>>(x, W, xh);

    node_att_kernel<<<((size_t)N * H * 32 + BT - 1) / BT, BT, 0, stream>>>(
        xh, att_src, att_dst, a_src, a_dst, N);
    edge_logits_kernel<<<(E + N + BT - 1) / BT, BT, 0, stream>>>(
        srcIdx, dstIdx, edge_attr, loop_at, vedge, a_src, a_dst, raw, amax, ei, E, N);
    edge_exp_kernel<<<((E + N) * H + BT - 1) / BT, BT, 0, stream>>>(
        dstIdx, raw, amax, asum, alpha, E, N);
    aggregate_kernel<<<((size_t)(E + N) * 32 + BT - 1) / BT, BT, 0, stream>>>(
        srcIdx, dstIdx, xh, asum, alpha, out, E, N);
}